// S4Conv1D_59459527246265
// MI455X (gfx1250) — compile-verified
//
#include <hip/hip_runtime.h>

// ---------------------------------------------------------------------------
// S4 recurrence y = C^T * scan(h' = Ad h + Bd x) over L = 2^22 = 2048*2048.
// Reformulated as y = <U_rev, Xc @ Vrev^T> with log-depth matrix doubling.
// Main GEMM: V_WMMA_F32_16X16X4_F32 (fp32 matrix pipe, wave32), tiles staged
// into LDS via global_load_async_to_lds_b128 (ASYNCcnt) when available.
// ---------------------------------------------------------------------------

typedef __attribute__((ext_vector_type(2))) float v2f;
typedef __attribute__((ext_vector_type(8))) float v8f;

static constexpr int    TCH     = 2048;         // chunk size (L = TCH*TCH)
static constexpr int    NKS     = 8;            // K-slices in main GEMM
// workspace float offsets
static constexpr size_t WS_AD   = 0;            // 64*64
static constexpr size_t WS_BD   = 4096;         // 64
static constexpr size_t WS_CT   = 4160;         // 64
static constexpr size_t WS_MPA  = 4224;         // 64*64  matrix-power buf A
static constexpr size_t WS_MPB  = 8320;         // 64*64  matrix-power buf B
static constexpr size_t WS_V    = 12416;        // 64*2048  V[k][j] = (Ad^j Bd)[k]
static constexpr size_t WS_U    = 143488;       // 2048*64  U[hi][k] = (C^T M^hi)[k]
static constexpr size_t WS_VR   = 274560;       // 64*2048  VR[k][i] = V[k][2047-i]
static constexpr size_t WS_WP   = 405632;       // 8*2048*64 partial W per K-slice

// --- CDNA5 async global->LDS staging (falls back to sync copy if absent) ---
#if defined(__AMDGCN__) && __has_builtin(__builtin_amdgcn_global_load_async_to_lds_b128)
#define S4_ASYNC_LDS 1
#else
#define S4_ASYNC_LDS 0
#endif

typedef int v4i __attribute__((vector_size(16)));
typedef __attribute__((address_space(1))) v4i as1_v4i;   // global
typedef __attribute__((address_space(3))) v4i as3_v4i;   // LDS

__device__ __forceinline__ void s4_copy16_to_lds(const void* g, void* l) {
#if S4_ASYNC_LDS
  __builtin_amdgcn_global_load_async_to_lds_b128(
      (as1_v4i*)(uintptr_t)g,
      (as3_v4i*)(unsigned int)(uintptr_t)l,   // LDS address = low 32 bits of the pointer (ISA 10.2)
      0, 0);
#else
  *(uint4*)l = *(const uint4*)g;
#endif
}

__device__ __forceinline__ void s4_async_wait0() {
#if S4_ASYNC_LDS
#if __has_builtin(__builtin_amdgcn_s_wait_asynccnt)
  __builtin_amdgcn_s_wait_asynccnt(0);
#else
  asm volatile("s_wait_asynccnt 0" ::: "memory");
#endif
#endif
}

// ---------------------------------------------------------------------------
// Setup: E = 2 - Lambda (elementwise), D = inv(E) (Gauss-Jordan, partial
// pivoting), A1 = D - (DP)(Q^T D)/(1 + Q^T D P), A0 = 2I + Lambda - P Q^T,
// Ad = A1 A0, Bd = 2 A1 B.  Seeds V[:,0]=Bd, U[0,:]=C, Mpow_A = Ad.
// ---------------------------------------------------------------------------
__global__ __launch_bounds__(256)
void s4_setup_kernel(const float* __restrict__ Lambda, const float* __restrict__ P,
                     const float* __restrict__ Q, const float* __restrict__ Bg,
                     const float* __restrict__ Cg, float* __restrict__ ws) {
  __shared__ float aug[64][130];   // [E | I] augmented, stride 130 vs bank conflicts
  __shared__ float Lm[64][64];
  __shared__ float Pv[64], Qv[64], Bl[64], Dp[64], qD[64], fr[64];
  __shared__ float sden;
  __shared__ int   piv;
  const int tid = threadIdx.x;

  for (int e = tid; e < 4096; e += 256) {
    int r = e >> 6, c = e & 63;
    float lv = Lambda[e];
    Lm[r][c]      = lv;
    aug[r][c]     = 2.0f - lv;
    aug[r][64+c]  = (r == c) ? 1.0f : 0.0f;
  }
  if (tid < 64) { Pv[tid] = P[tid]; Qv[tid] = Q[tid]; Bl[tid] = Bg[tid]; }
  __syncthreads();

  // Gauss-Jordan with partial pivoting
  for (int k = 0; k < 64; ++k) {
    if (tid == 0) {
      int best = k; float bv = fabsf(aug[k][k]);
      for (int j = k + 1; j < 64; ++j) {
        float v = fabsf(aug[j][k]);
        if (v > bv) { bv = v; best = j; }
      }
      piv = best;
    }
    __syncthreads();
    const int pr = piv;
    if (pr != k && tid < 128) {
      float t = aug[k][tid]; aug[k][tid] = aug[pr][tid]; aug[pr][tid] = t;
    }
    __syncthreads();
    const float pvv = aug[k][k];
    __syncthreads();
    if (tid < 128) aug[k][tid] = aug[k][tid] / pvv;     // normalize pivot row
    if (tid < 64 && tid != k) fr[tid] = aug[tid][k];    // elimination factors
    __syncthreads();
    {
      int r = tid >> 2, c0 = (tid & 3) << 5;            // 4 threads per row
      if (r != k) {
        float f = fr[r];
        #pragma unroll 8
        for (int c = c0; c < c0 + 32; ++c) aug[r][c] -= f * aug[k][c];
      }
    }
    __syncthreads();
  }

  // Dp = D P, qD = Q^T D, sden = 1 + Q^T D P      (D = right half of aug)
  if (tid < 64) {
    float s = 0.0f, s2 = 0.0f;
    for (int c = 0; c < 64; ++c) s  += aug[tid][64 + c] * Pv[c];
    for (int r = 0; r < 64; ++r) s2 += Qv[r] * aug[r][64 + tid];
    Dp[tid] = s; qD[tid] = s2;
  }
  __syncthreads();
  if (tid == 0) {
    float s = 1.0f;
    for (int c = 0; c < 64; ++c) s += qD[c] * Pv[c];
    sden = s;
  }
  __syncthreads();
  // A1 into left half of aug
  for (int e = tid; e < 4096; e += 256) {
    int r = e >> 6, c = e & 63;
    aug[r][c] = aug[r][64 + c] - Dp[r] * qD[c] / sden;
  }
  __syncthreads();
  // Ad = A1 @ A0,  A0[t][c] = 2*(t==c) + Lm[t][c] - Pv[t]*Qv[c]
  for (int e = tid; e < 4096; e += 256) {
    int r = e >> 6, c = e & 63;
    float s = 0.0f;
    for (int t = 0; t < 64; ++t) {
      float a0 = Lm[t][c] - Pv[t] * Qv[c];
      if (t == c) a0 += 2.0f;
      s += aug[r][t] * a0;
    }
    ws[WS_AD  + e] = s;
    ws[WS_MPA + e] = s;          // Mpow buffer A starts at Ad^1
  }
  if (tid < 64) {
    float s = 0.0f;
    for (int t = 0; t < 64; ++t) s += aug[tid][t] * Bl[t];
    float bd = 2.0f * s;
    ws[WS_BD + tid] = bd;
    ws[WS_V + (size_t)tid * TCH] = bd;   // V[:,0] = Bd
    float cv = Cg[tid];
    ws[WS_CT + tid] = cv;
    ws[WS_U + tid]  = cv;                // U[0,:] = C^T
  }
}

// ---------------------------------------------------------------------------
// V doubling: cols [m,2m) = Mpow(src) @ V[:,0:m); last block squares Mpow.
// srcSel: 0 -> src MPA / dst MPB, 1 -> src MPB / dst MPA.
// ---------------------------------------------------------------------------
__global__ __launch_bounds__(256)
void s4_vstep_kernel(float* __restrict__ ws, int m, int srcSel) {
  const float* M  = ws + (srcSel ? WS_MPB : WS_MPA);
  float*       Mn = ws + (srcSel ? WS_MPA : WS_MPB);
  float*       V  = ws + WS_V;
  const int tid = threadIdx.x;
  if (blockIdx.x == gridDim.x - 1) {            // square Mpow
    for (int o = tid; o < 4096; o += 256) {
      int r = o >> 6, c = o & 63;
      float s = 0.0f;
      for (int t = 0; t < 64; ++t) s += M[r*64 + t] * M[t*64 + c];
      Mn[o] = s;
    }
  } else {
    int gid = blockIdx.x * 256 + tid;
    if (gid < 64 * m) {
      int k = gid / m, j = gid - k * m;
      float s = 0.0f;
      for (int t = 0; t < 64; ++t) s += M[k*64 + t] * V[(size_t)t*TCH + j];
      V[(size_t)k*TCH + m + j] = s;
    }
  }
}

// U doubling: rows [m,2m) = U[0:m) @ Npow(src); last block squares Npow.
__global__ __launch_bounds__(256)
void s4_ustep_kernel(float* __restrict__ ws, int m, int srcSel) {
  const float* M  = ws + (srcSel ? WS_MPB : WS_MPA);
  float*       Mn = ws + (srcSel ? WS_MPA : WS_MPB);
  float*       U  = ws + WS_U;
  const int tid = threadIdx.x;
  if (blockIdx.x == gridDim.x - 1) {
    for (int o = tid; o < 4096; o += 256) {
      int r = o >> 6, c = o & 63;
      float s = 0.0f;
      for (int t = 0; t < 64; ++t) s += M[r*64 + t] * M[t*64 + c];
      Mn[o] = s;
    }
  } else {
    int gid = blockIdx.x * 256 + tid;
    if (gid < 64 * m) {
      int j = gid >> 6, k = gid & 63;
      float s = 0.0f;
      for (int t = 0; t < 64; ++t) s += U[(size_t)j*64 + t] * M[t*64 + k];
      U[(size_t)(m + j)*64 + k] = s;
    }
  }
}

// VR[k][i] = V[k][2047 - i]  (row reversal; keeps K contiguous per output col
// so the GEMM B-tile stages as aligned b128 DMAs and reads as ds_load_b64)
__global__ __launch_bounds__(256)
void s4_vrev_kernel(float* __restrict__ ws) {
  int idx = blockIdx.x * 256 + threadIdx.x;      // 512 blocks * 256 = 131072
  int k = idx >> 11, i = idx & 2047;
  ws[WS_VR + (size_t)k * TCH + i] = ws[WS_V + (size_t)k * TCH + (TCH - 1 - i)];
}

// ---------------------------------------------------------------------------
// Main GEMM: Wp[ks] = Xc(2048x2048) @ VR^T(2048x64) over K-slice ks.
// grid = 32 M-blocks x 8 K-slices, 128 threads (4 waves), wave = 16x64 tile.
// fp32 WMMA 16x16x4; 64x64 tiles staged via async b128 DMA into LDS
// (row stride 68 floats = 272 B: 16B-aligned DMA targets, conflict-free
// b64 fragment reads with bank stride 4).
// ---------------------------------------------------------------------------
__global__ __launch_bounds__(128)
void s4_gemm_kernel(const float* __restrict__ x, float* __restrict__ ws) {
  __shared__ float Xt[64][68];   // Xt[row][k]   (A operand, K contiguous)
  __shared__ float Vt[64][68];   // Vt[ncol][k]  (B operand, K contiguous)
  const float* vr = ws + WS_VR;
  float*       wp = ws + WS_WP;
  const int tid  = threadIdx.x;
  const int wave = tid >> 5, lane = tid & 31;
  const int mblk = blockIdx.x & 31;              // 32 blocks of 64 rows
  const int ks   = blockIdx.x >> 5;              // 8 K-slices of 256
  const int kbase = ks * (TCH / NKS);
  const int mcol = lane & 15;                    // M row (A) / N col (B,D)
  const int hh   = lane >> 4;                    // half-wave select

  v8f acc[4];
  const v8f vzero = {0.f,0.f,0.f,0.f,0.f,0.f,0.f,0.f};
  #pragma unroll
  for (int i = 0; i < 4; ++i) acc[i] = vzero;

  for (int kc = 0; kc < TCH / NKS; kc += 64) {
    __syncthreads();                             // LDS reuse (reads done pre-WMMA)
    for (int q = tid; q < 1024; q += 128) {      // 64 rows x 16 b128 chunks
      int r = q >> 4, c4 = (q & 15) << 2;
      s4_copy16_to_lds(&x[(size_t)(mblk*64 + r) * TCH + (kbase + kc + c4)],
                       &Xt[r][c4]);
      s4_copy16_to_lds(&vr[(size_t)r * TCH + (kbase + kc + c4)],
                       &Vt[r][c4]);
    }
    if (kc + 64 < TCH / NKS) {                   // global_prefetch_b8 next X tile
      __builtin_prefetch(&x[(size_t)(mblk*64 + (tid >> 1)) * TCH +
                            (kbase + kc + 64 + (tid & 1) * 32)], 0, 1);
    }
    s4_async_wait0();                            // s_wait_asynccnt 0
    __syncthreads();
    #pragma unroll
    for (int kk = 0; kk < 64; kk += 4) {
      const int kh = kk + hh * 2;
      v2f a = { Xt[wave*16 + mcol][kh], Xt[wave*16 + mcol][kh + 1] };
      #pragma unroll
      for (int nt = 0; nt < 4; ++nt) {
        v2f b = { Vt[nt*16 + mcol][kh], Vt[nt*16 + mcol][kh + 1] };
        acc[nt] = __builtin_amdgcn_wmma_f32_16x16x4_f32(
            false, a, false, b, (short)0, acc[nt], false, false);
      }
    }
  }
  // D layout: VGPR r holds M = r + 8*hh, N = lane&15
  #pragma unroll
  for (int nt = 0; nt < 4; ++nt)
    #pragma unroll
    for (int r = 0; r < 8; ++r) {
      int row = mblk*64 + wave*16 + r + 8*hh;
      int col = nt*16 + mcol;
      wp[((size_t)ks * TCH + row) * 64 + col] = acc[nt][r];
    }
}

// y = sum_{c,k} U[2047-c][k] * sum_s Wp[s][c][k]   (single block, deterministic)
__global__ __launch_bounds__(256)
void s4_reduce_kernel(const float* __restrict__ ws, float* __restrict__ out) {
  __shared__ float red[256];
  const int tid = threadIdx.x;
  float p = 0.0f;
  for (int idx = tid; idx < TCH * 64; idx += 256) {
    int c = idx >> 6, k = idx & 63;
    float w = 0.0f;
    #pragma unroll
    for (int s = 0; s < NKS; ++s) w += ws[WS_WP + (size_t)s * (TCH * 64) + idx];
    p += ws[WS_U + (size_t)(TCH - 1 - c) * 64 + k] * w;
  }
  red[tid] = p;
  __syncthreads();
  for (int off = 128; off > 0; off >>= 1) {
    if (tid < off) red[tid] += red[tid + off];
    __syncthreads();
  }
  if (tid == 0) out[0] = red[0];
}

extern "C" void kernel_launch(void* const* d_in, const int* in_sizes, int n_in,
                              void* d_out, int out_size, void* d_ws, size_t ws_size,
                              hipStream_t stream) {
  (void)in_sizes; (void)n_in; (void)out_size; (void)ws_size;
  const float* x      = (const float*)d_in[0];
  const float* Lambda = (const float*)d_in[1];
  const float* P      = (const float*)d_in[2];
  const float* Q      = (const float*)d_in[3];
  const float* B      = (const float*)d_in[4];
  const float* C      = (const float*)d_in[5];
  float* ws  = (float*)d_ws;
  float* out = (float*)d_out;

  s4_setup_kernel<<<1, 256, 0, stream>>>(Lambda, P, Q, B, C, ws);
  // V: columns Ad^j Bd, j in [0,2048) via 11 doublings (Mpow starts in MPA)
  for (int i = 0; i < 11; ++i) {
    int m  = 1 << i;
    int nb = (64 * m + 255) / 256;
    s4_vstep_kernel<<<nb + 1, 256, 0, stream>>>(ws, m, i & 1);
  }
  // U: rows C^T M^hi, M = Ad^2048 (ends in MPB after 11 V-steps)
  for (int i = 0; i < 11; ++i) {
    int m  = 1 << i;
    int nb = (64 * m + 255) / 256;
    s4_ustep_kernel<<<nb + 1, 256, 0, stream>>>(ws, m, (i & 1) ^ 1);
  }
  s4_vrev_kernel<<<512, 256, 0, stream>>>(ws);
  s4_gemm_kernel<<<32 * NKS, 128, 0, stream>>>(x, ws);
  s4_reduce_kernel<<<1, 256, 0, stream>>>(ws, out);
}